// RC_38706245271596
// MI455X (gfx1250) — compile-verified
//
#include <hip/hip_runtime.h>
#include <hip/hip_bf16.h>

#define BB   256
#define TT   300
#define DIN  700
#define HIDN 128
#define OUTD 20
#define NL   4
#define MROWS (BB*TT)

typedef __attribute__((ext_vector_type(2))) float        v2f;
typedef __attribute__((ext_vector_type(8))) float        v8f;
typedef __attribute__((ext_vector_type(4))) unsigned int u32x4;
typedef __attribute__((ext_vector_type(8))) int          i32x8;
typedef __attribute__((ext_vector_type(4))) int          i32x4;

// ---------------------------------------------------------------------------
// Tensor Data Mover: 2D tile load Global -> LDS (f32 elements).
//   tile_w  : row length in elements (contiguous in memory and in LDS)
//   tile_h  : number of rows
//   rstride : row stride in ELEMENTS (D# strides are in data_size units)
// D# built per CDNA5 ISA (group0 bits: count=1, lds_addr
// [63:32], global_addr [120:64], type=2 [127:126]; group1: data_size=4B,
// tensor dims == tile dims so OOB never triggers, dim0 stride).
// This toolchain exposes the 6-arg builtin (g0, g1, g2, g3, g4, cpol).
// Issue from ONE wave only; completion via TENSORcnt.
// ---------------------------------------------------------------------------
__device__ __forceinline__ void tdm_load_2d(const void* lds_ptr, const void* gptr,
                                            unsigned tile_w, unsigned tile_h,
                                            unsigned long long rstride) {
    const unsigned           lds_off = (unsigned)(unsigned long long)(uintptr_t)lds_ptr;
    const unsigned long long ga      = (unsigned long long)(uintptr_t)gptr;
    u32x4 g0;
    g0[0] = 1u;                                               // count=1 (valid), user mode
    g0[1] = lds_off;                                          // lds_addr
    g0[2] = (unsigned)ga;                                     // global_addr[31:0]
    g0[3] = (unsigned)((ga >> 32) & 0x01ffffffu) | (2u << 30);// global_addr[56:32] | type=2
    i32x8 g1;
    g1[0] = (int)(2u << 16);                                  // wg_mask=0, data_size=4B
    g1[1] = (int)((tile_w & 0xffffu) << 16);                  // tensor_dim0[15:0]
    g1[2] = (int)(((tile_w >> 16) & 0xffffu) | ((tile_h & 0xffffu) << 16)); // dim0 hi | dim1 lo
    g1[3] = (int)(((tile_h >> 16) & 0xffffu) | ((tile_w & 0xffffu) << 16)); // dim1 hi | tile_dim0
    g1[4] = (int)(tile_h & 0xffffu);                          // tile_dim1 (tile_dim2=0)
    g1[5] = (int)(unsigned)rstride;                           // tensor_dim0_stride[31:0]
    g1[6] = (int)((unsigned)(rstride >> 32) & 0xffffu);       // dim0_stride[47:32]
    g1[7] = 0;
    i32x4 z4 = {0, 0, 0, 0};
    i32x8 z8 = {0, 0, 0, 0, 0, 0, 0, 0};
    __builtin_amdgcn_tensor_load_to_lds(g0, g1, z4, z4, z8, 0);
}

// ---------------------------------------------------------------------------
// Kernel 1: inp = x @ w_in + b_in   (M=76800, K=700, N=128)
// 8 waves, each owns a 16x16 f32 WMMA C tile. w_in K-chunks (64x128) stream
// through double-buffered LDS via TDM, overlapped with the WMMA chain.
// ---------------------------------------------------------------------------
__global__ __launch_bounds__(256)
void k_inp_gemm(const float* __restrict__ x, const float* __restrict__ w_in,
                const float* __restrict__ b_in, float* __restrict__ inp) {
    __shared__ float lA[2][16 * 64];
    __shared__ float lB[2][64 * 128];
    const int row0 = blockIdx.x * 16;
    const int tid  = threadIdx.x;
    const int wave = tid >> 5;
    const int lane = tid & 31;
    const int hi   = lane >> 4;
    const int lo   = lane & 15;
    const int n0   = wave * 16;
    const int nchunks = (DIN + 63) / 64;        // 11 (last chunk = 60 rows)

    // --- preload chunk 0 ---
    if (wave == 0)
        tdm_load_2d(&lB[0][0], w_in, 128, 64, HIDN);
    for (int i = tid; i < 16 * 64; i += 256) {
        int r = i >> 6, k = i & 63;
        lA[0][i] = x[(size_t)(row0 + r) * DIN + k];
    }
    if (wave == 0) __builtin_amdgcn_s_wait_tensorcnt(0);
    __syncthreads();

    v8f c = {};
    for (int ci = 0; ci < nchunks; ++ci) {
        const int cur = ci & 1, nxt = cur ^ 1;
        const int kcn = (ci + 1) * 64;
        if (ci + 1 < nchunks) {
            const int kh = (DIN - kcn > 64) ? 64 : (DIN - kcn);
            if (wave == 0)                      // async DMA of next w_in chunk
                tdm_load_2d(&lB[nxt][0], w_in + (size_t)kcn * HIDN, 128, kh, HIDN);
            for (int i = tid; i < 16 * 64; i += 256) {   // next x tile (guarded tail)
                int r = i >> 6, k = i & 63;
                int kg = kcn + k;
                lA[nxt][i] = (kg < DIN) ? x[(size_t)(row0 + r) * DIN + kg] : 0.f;
            }
            if (kh < 64)                        // zero stale tail rows (disjoint from TDM)
                for (int i = tid; i < (64 - kh) * 128; i += 256)
                    lB[nxt][kh * 128 + i] = 0.f;
        }
        // --- 16 WMMA on current buffers ---
        #pragma unroll
        for (int kk = 0; kk < 64; kk += 4) {
            const int ka = kk + 2 * hi;
            v2f av, bv;
            av.x = lA[cur][lo * 64 + ka];
            av.y = lA[cur][lo * 64 + ka + 1];
            bv.x = lB[cur][ka * 128 + n0 + lo];
            bv.y = lB[cur][(ka + 1) * 128 + n0 + lo];
            c = __builtin_amdgcn_wmma_f32_16x16x4_f32(false, av, false, bv,
                                                      (short)0, c, false, false);
        }
        if (wave == 0 && ci + 1 < nchunks) __builtin_amdgcn_s_wait_tensorcnt(0);
        __syncthreads();
    }
    const int col  = n0 + lo;
    const float bi = b_in[col];
    #pragma unroll
    for (int r = 0; r < 8; ++r) {
        int row = row0 + r + 8 * hi;            // WMMA C layout: M = r (+8 for hi half)
        inp[(size_t)row * HIDN + col] = c[r] + bi;
    }
}

// ---------------------------------------------------------------------------
// Kernel 2: fused drive-GEMM + SNN scan per layer.
// A_w[l] (64KB) TDM-loaded once into LDS; the per-timestep 16x128 inp tile is
// double-buffered via TDM (tile t+1 in flight behind the WMMA chain of t).
// mem/spk/spike-sum live in WMMA C-layout registers for all T=300 steps.
// ---------------------------------------------------------------------------
__global__ __launch_bounds__(256)
void k_reservoir_scan(const float* __restrict__ inp, const float* __restrict__ A_w,
                      const float* __restrict__ A_b, const float* __restrict__ thr,
                      const float* __restrict__ decay, const float* __restrict__ rstv,
                      const float* __restrict__ mem0, float* __restrict__ rates,
                      float* __restrict__ hid4_mem, float* __restrict__ hid4_spk) {
    __shared__ float lW[128 * 128];             // 64KB: A_w[l]
    __shared__ float lX[2][16 * 128];           // 2x8KB: inp tiles (double buffer)
    const int l   = blockIdx.y;
    const int b0  = blockIdx.x * 16;
    const int tid = threadIdx.x;
    const int wave = tid >> 5;
    const int lane = tid & 31;
    const int hi = lane >> 4, lo = lane & 15;
    const int n0 = wave * 16;
    const int col = n0 + lo;

    if (wave == 0) {
        tdm_load_2d(&lW[0], A_w + (size_t)l * HIDN * HIDN, 128, 128, HIDN);
        tdm_load_2d(&lX[0][0], inp + (size_t)b0 * TT * HIDN, 128, 16,
                    (unsigned long long)TT * HIDN);
        __builtin_amdgcn_s_wait_tensorcnt(0);   // TENSORcnt in-order per wave
    }
    __syncthreads();

    const float th = thr[col];
    const float dc = decay[col];
    const float rs = rstv[col];
    const float ab = A_b[l * HIDN + col];

    float mem[8], spk[8], ssum[8];
    #pragma unroll
    for (int r = 0; r < 8; ++r) {
        const int b = b0 + r + 8 * hi;
        mem[r]  = mem0[((size_t)l * BB + b) * HIDN + col];
        spk[r]  = 0.f;
        ssum[r] = 0.f;
    }
    if (l == 3) {                               // t=0 rows of the traces
        #pragma unroll
        for (int r = 0; r < 8; ++r) {
            const int b = b0 + r + 8 * hi;
            hid4_mem[((size_t)b * (TT + 1)) * HIDN + col] = mem[r];
            hid4_spk[((size_t)b * (TT + 1)) * HIDN + col] = 0.f;
        }
    }

    for (int t = 0; t < TT; ++t) {
        const float* lXc = lX[t & 1];
        if (wave == 0 && t + 1 < TT)            // DMA tile t+1 behind this step's math
            tdm_load_2d(&lX[(t + 1) & 1][0],
                        inp + ((size_t)b0 * TT + (t + 1)) * HIDN, 128, 16,
                        (unsigned long long)TT * HIDN);

        v8f c;
        #pragma unroll
        for (int r = 0; r < 8; ++r) c[r] = ab;

        #pragma unroll 4
        for (int kk = 0; kk < 128; kk += 4) {
            const int ka = kk + 2 * hi;
            v2f av, bv;
            av.x = lXc[lo * 128 + ka];
            av.y = lXc[lo * 128 + ka + 1];
            bv.x = lW[ka * 128 + col];
            bv.y = lW[(ka + 1) * 128 + col];
            c = __builtin_amdgcn_wmma_f32_16x16x4_f32(false, av, false, bv,
                                                      (short)0, c, false, false);
        }

        #pragma unroll
        for (int r = 0; r < 8; ++r) {
            const float m = rs * spk[r] + mem[r] * dc * (1.f - spk[r]) + c[r];
            const float s = (m - th > 0.f) ? 1.f : 0.f;
            mem[r] = m; spk[r] = s; ssum[r] += s;
            if (l == 3) {
                const int b = b0 + r + 8 * hi;
                hid4_mem[((size_t)b * (TT + 1) + (t + 1)) * HIDN + col] = m;
                hid4_spk[((size_t)b * (TT + 1) + (t + 1)) * HIDN + col] = s;
            }
        }
        if (wave == 0 && t + 1 < TT) __builtin_amdgcn_s_wait_tensorcnt(0);
        __syncthreads();
    }

    #pragma unroll
    for (int r = 0; r < 8; ++r) {
        const int b = b0 + r + 8 * hi;
        rates[((size_t)l * BB + b) * HIDN + col] = ssum[r] * (1.f / (float)TT);
    }
}

// ---------------------------------------------------------------------------
// Kernel 3: cat[b, l*128+n] = relu(rates[l] @ fc_w[l] + fc_b[l])
// ---------------------------------------------------------------------------
__global__ __launch_bounds__(256)
void k_fc(const float* __restrict__ rates, const float* __restrict__ fc_w,
          const float* __restrict__ fc_b, float* __restrict__ catbuf) {
    __shared__ float lW[128 * 128];
    __shared__ float lX[16 * 128];
    const int l  = blockIdx.y;
    const int b0 = blockIdx.x * 16;
    const int tid = threadIdx.x;
    const int wave = tid >> 5;
    const int lane = tid & 31;
    const int hi = lane >> 4, lo = lane & 15;
    const int n0 = wave * 16;
    const int col = n0 + lo;

    if (wave == 0) {
        tdm_load_2d(&lW[0], fc_w + (size_t)l * HIDN * HIDN, 128, 128, HIDN);
        tdm_load_2d(&lX[0], rates + ((size_t)l * BB + b0) * HIDN, 128, 16, HIDN);
        __builtin_amdgcn_s_wait_tensorcnt(0);
    }
    __syncthreads();

    v8f c;
    const float fb = fc_b[l * HIDN + col];
    #pragma unroll
    for (int r = 0; r < 8; ++r) c[r] = fb;

    #pragma unroll 4
    for (int kk = 0; kk < 128; kk += 4) {
        const int ka = kk + 2 * hi;
        v2f av, bv;
        av.x = lX[lo * 128 + ka];
        av.y = lX[lo * 128 + ka + 1];
        bv.x = lW[ka * 128 + col];
        bv.y = lW[(ka + 1) * 128 + col];
        c = __builtin_amdgcn_wmma_f32_16x16x4_f32(false, av, false, bv,
                                                  (short)0, c, false, false);
    }

    #pragma unroll
    for (int r = 0; r < 8; ++r) {
        const int b = b0 + r + 8 * hi;
        const float v = c[r];
        catbuf[(size_t)b * (NL * HIDN) + l * HIDN + col] = v > 0.f ? v : 0.f;
    }
}

// ---------------------------------------------------------------------------
// Kernel 4: out = cat @ w_out + b_out  ([256,512]x[512,20] — tiny)
// ---------------------------------------------------------------------------
__global__ __launch_bounds__(256)
void k_out(const float* __restrict__ catbuf, const float* __restrict__ w_out,
           const float* __restrict__ b_out, float* __restrict__ out) {
    const int idx = blockIdx.x * 256 + threadIdx.x;
    if (idx >= BB * OUTD) return;
    const int b = idx / OUTD, o = idx % OUTD;
    float s = b_out[o];
    for (int k = 0; k < NL * HIDN; ++k)
        s += catbuf[(size_t)b * (NL * HIDN) + k] * w_out[(size_t)k * OUTD + o];
    out[idx] = s;
}

// ---------------------------------------------------------------------------
// Kernel 5: A_norm = sum |A_w|  (65536 elems, one block)
// ---------------------------------------------------------------------------
__global__ __launch_bounds__(256)
void k_anorm(const float* __restrict__ A_w, float* __restrict__ outp) {
    __shared__ float red[256];
    float s = 0.f;
    for (int i = threadIdx.x; i < NL * HIDN * HIDN; i += 256)
        s += fabsf(A_w[i]);
    red[threadIdx.x] = s;
    __syncthreads();
    for (int off = 128; off > 0; off >>= 1) {
        if (threadIdx.x < off) red[threadIdx.x] += red[threadIdx.x + off];
        __syncthreads();
    }
    if (threadIdx.x == 0) *outp = red[0];
}

extern "C" void kernel_launch(void* const* d_in, const int* in_sizes, int n_in,
                              void* d_out, int out_size, void* d_ws, size_t ws_size,
                              hipStream_t stream) {
    const float* x     = (const float*)d_in[0];
    const float* w_in  = (const float*)d_in[1];
    const float* b_in  = (const float*)d_in[2];
    const float* A_w   = (const float*)d_in[3];
    const float* A_b   = (const float*)d_in[4];
    const float* fc_w  = (const float*)d_in[5];
    const float* fc_b  = (const float*)d_in[6];
    const float* w_out = (const float*)d_in[7];
    const float* b_out = (const float*)d_in[8];
    const float* thr   = (const float*)d_in[9];
    const float* decay = (const float*)d_in[10];
    const float* rstv  = (const float*)d_in[11];
    const float* mem0  = (const float*)d_in[12];

    float* out      = (float*)d_out;                       // [256,20]
    float* hid4_mem = out + (size_t)BB * OUTD;             // [256,301,128]
    float* hid4_spk = hid4_mem + (size_t)BB * (TT + 1) * HIDN;
    float* anorm    = hid4_spk + (size_t)BB * (TT + 1) * HIDN;

    float* ws       = (float*)d_ws;
    float* ws_inp   = ws;                                  // [76800,128] = 39.3MB
    float* ws_rates = ws_inp + (size_t)MROWS * HIDN;       // [4,256,128]
    float* ws_cat   = ws_rates + (size_t)NL * BB * HIDN;   // [256,512]

    k_inp_gemm<<<dim3(MROWS / 16), 256, 0, stream>>>(x, w_in, b_in, ws_inp);
    k_reservoir_scan<<<dim3(BB / 16, NL), 256, 0, stream>>>(
        ws_inp, A_w, A_b, thr, decay, rstv, mem0, ws_rates, hid4_mem, hid4_spk);
    k_fc<<<dim3(BB / 16, NL), 256, 0, stream>>>(ws_rates, fc_w, fc_b, ws_cat);
    k_out<<<(BB * OUTD + 255) / 256, 256, 0, stream>>>(ws_cat, w_out, b_out, out);
    k_anorm<<<1, 256, 0, stream>>>(A_w, anorm);
}